// GraphTransformer_34978213659049
// MI455X (gfx1250) — compile-verified
//
#include <hip/hip_runtime.h>
#include <stdint.h>

// -------------------------------------------------------------------------
// GraphTransformer layer for MI455X (gfx1250, wave32).
//   q/k/v projections hoisted from edges (E=800k) to nodes (N=50k): 16x fewer
//   GEMM flops; Q/K/V (77MB) stay L2-resident (192MB L2) for edge gathers.
//   GEMMs use V_WMMA_F32_16X16X4_F32 (fp32-exact vs the f32 reference),
//   M-tiled 64 rows/block so each B-fragment load feeds 4 WMMAs.
//   Tiles staged to LDS with global_load_async_to_lds_b128 (ASYNCcnt).
//   Edge messages head-reduced in LDS before atomicAdd (4x fewer atomics).
//   Full-tile blocks take straight-line, guard-free store paths.
// -------------------------------------------------------------------------

typedef __attribute__((ext_vector_type(2))) float v2f;
typedef __attribute__((ext_vector_type(8))) float v8f;

#define D_FEAT   128
#define D_EDGE   64
#define HID      128
#define OUT_DIM  32
#define N_HEADS  4
#define MT       4     // M-tiles (of 16 rows) per block -> 64 rows/block

__device__ __forceinline__ v8f wmma4(v2f a, v2f b, v8f c) {
  // (neg_a, A, neg_b, B, c_mod, C, reuse_a, reuse_b)
  return __builtin_amdgcn_wmma_f32_16x16x4_f32(false, a, false, b, (short)0, c,
                                               false, false);
}

// Monotone float<->uint encoding so float max == unsigned atomicMax.
__device__ __forceinline__ unsigned fenc(float f) {
  unsigned u = __float_as_uint(f);
  return (u & 0x80000000u) ? ~u : (u | 0x80000000u);
}
__device__ __forceinline__ float fdec(unsigned u) {
  unsigned v = (u & 0x80000000u) ? (u & 0x7FFFFFFFu) : ~u;
  return __uint_as_float(v);
}
#define ENC_NEG_INF 0x007FFFFFu  // fenc(-inf)

// ---------------------------------------------------------------- K0: init
__global__ __launch_bounds__(256) void k_init(float* __restrict__ aggr,
                                              float* __restrict__ nodeSum,
                                              unsigned* __restrict__ nodeMax,
                                              int nn) {
  int t = blockIdx.x * blockDim.x + threadIdx.x;
  if (t < nn * OUT_DIM) aggr[t] = 0.0f;
  if (t < nn * N_HEADS) { nodeSum[t] = 0.0f; nodeMax[t] = ENC_NEG_INF; }
}

// ------------------------------------------ K1: node projections (WMMA f32)
// 64 nodes per block (4 m-tiles), 8 waves; wave w computes HID columns
// [16w,16w+16) of Q,K,V; waves 0..1 then compute feats = nodes@Wu + bu.
#define LDSA_STRIDE 132  // 128 + 4 pad -> conflict-spread A reads
__global__ __launch_bounds__(256) void k_node_proj(
    const float* __restrict__ nodes,
    const float* __restrict__ Wq, const float* __restrict__ bq,
    const float* __restrict__ Wk, const float* __restrict__ bk,
    const float* __restrict__ Wv, const float* __restrict__ bv,
    const float* __restrict__ Wu, const float* __restrict__ bu,
    float* __restrict__ Q, float* __restrict__ Kp, float* __restrict__ Vp,
    float* __restrict__ feats, int nn) {
  __shared__ float As[16 * MT * LDSA_STRIDE];  // 64 x 132 f32 = 33.8 KB
  const int tid = threadIdx.x;
  const int nb  = blockIdx.x * (16 * MT);
  const bool full = (nb + 16 * MT) <= nn;      // block-uniform tail test

  // Stage 64x128 fp32 node tile into LDS: 128B/thread via 8 async b128.
  {
    int r    = tid >> 2;                      // tile row 0..63
    int grow = nb + r; if (grow >= nn) grow = nn - 1;  // clamp (reads only)
    int cf   = (tid & 3) * 32;                // first float within row
    uint32_t lds = (uint32_t)(uintptr_t)(&As[r * LDSA_STRIDE + cf]);
    unsigned long long g =
        (unsigned long long)(uintptr_t)(nodes + (size_t)grow * D_FEAT + cf);
    asm volatile(
        "global_load_async_to_lds_b128 %0, %1, off\n\t"
        "global_load_async_to_lds_b128 %0, %1, off offset:16\n\t"
        "global_load_async_to_lds_b128 %0, %1, off offset:32\n\t"
        "global_load_async_to_lds_b128 %0, %1, off offset:48\n\t"
        "global_load_async_to_lds_b128 %0, %1, off offset:64\n\t"
        "global_load_async_to_lds_b128 %0, %1, off offset:80\n\t"
        "global_load_async_to_lds_b128 %0, %1, off offset:96\n\t"
        "global_load_async_to_lds_b128 %0, %1, off offset:112\n\t"
        "s_wait_asynccnt 0"
        :: "v"(lds), "v"(g) : "memory");
  }
  __syncthreads();

  const int wave = tid >> 5;
  const int lane = tid & 31;
  const int col  = lane & 15;   // A row (within tile) == B/C column
  const int half = lane >> 4;
  const int ksel = half * 2;
  const int cw   = wave * 16;

  v8f cq[MT] = {}, ck[MT] = {}, cv[MT] = {};
  for (int kk = 0; kk < D_FEAT / 4; ++kk) {
    const int k0 = kk * 4 + ksel;
    v2f a[MT];
#pragma unroll
    for (int mt = 0; mt < MT; ++mt) {
      a[mt].x = As[(mt * 16 + col) * LDSA_STRIDE + k0];
      a[mt].y = As[(mt * 16 + col) * LDSA_STRIDE + k0 + 1];
    }
    v2f b;
    b.x = Wq[(size_t)k0 * HID + cw + col];
    b.y = Wq[(size_t)(k0 + 1) * HID + cw + col];
#pragma unroll
    for (int mt = 0; mt < MT; ++mt) cq[mt] = wmma4(a[mt], b, cq[mt]);
    b.x = Wk[(size_t)k0 * HID + cw + col];
    b.y = Wk[(size_t)(k0 + 1) * HID + cw + col];
#pragma unroll
    for (int mt = 0; mt < MT; ++mt) ck[mt] = wmma4(a[mt], b, ck[mt]);
    b.x = Wv[(size_t)k0 * HID + cw + col];
    b.y = Wv[(size_t)(k0 + 1) * HID + cw + col];
#pragma unroll
    for (int mt = 0; mt < MT; ++mt) cv[mt] = wmma4(a[mt], b, cv[mt]);
  }

  const float bqv = bq[cw + col], bkv = bk[cw + col], bvv = bv[cw + col];
  const int rbase = half * 8;
  if (full) {  // straight-line stores, no per-element exec masking
#pragma unroll
    for (int mt = 0; mt < MT; ++mt) {
#pragma unroll
      for (int i = 0; i < 8; ++i) {
        const int node = nb + mt * 16 + rbase + i;
        Q [(size_t)node * HID + cw + col] = cq[mt][i] + bqv;
        Kp[(size_t)node * HID + cw + col] = ck[mt][i] + bkv;
        Vp[(size_t)node * HID + cw + col] = cv[mt][i] + bvv;
      }
    }
  } else {
#pragma unroll
    for (int mt = 0; mt < MT; ++mt) {
#pragma unroll
      for (int i = 0; i < 8; ++i) {
        const int node = nb + mt * 16 + rbase + i;
        if (node < nn) {
          Q [(size_t)node * HID + cw + col] = cq[mt][i] + bqv;
          Kp[(size_t)node * HID + cw + col] = ck[mt][i] + bkv;
          Vp[(size_t)node * HID + cw + col] = cv[mt][i] + bvv;
        }
      }
    }
  }

  // feats = nodes @ Wu + bu : only waves 0..1 (OUT_DIM = 32 columns).
  // Wave-uniform branch -> EXEC remains all-ones for the WMMAs inside.
  if (wave < 2) {
    v8f cu[MT] = {};
    for (int kk = 0; kk < D_FEAT / 4; ++kk) {
      const int k0 = kk * 4 + ksel;
      v2f b;
      b.x = Wu[(size_t)k0 * OUT_DIM + cw + col];
      b.y = Wu[(size_t)(k0 + 1) * OUT_DIM + cw + col];
#pragma unroll
      for (int mt = 0; mt < MT; ++mt) {
        v2f a;
        a.x = As[(mt * 16 + col) * LDSA_STRIDE + k0];
        a.y = As[(mt * 16 + col) * LDSA_STRIDE + k0 + 1];
        cu[mt] = wmma4(a, b, cu[mt]);
      }
    }
    const float buv = bu[cw + col];
    if (full) {
#pragma unroll
      for (int mt = 0; mt < MT; ++mt) {
#pragma unroll
        for (int i = 0; i < 8; ++i) {
          const int node = nb + mt * 16 + rbase + i;
          feats[(size_t)node * OUT_DIM + cw + col] = cu[mt][i] + buv;
        }
      }
    } else {
#pragma unroll
      for (int mt = 0; mt < MT; ++mt) {
#pragma unroll
        for (int i = 0; i < 8; ++i) {
          const int node = nb + mt * 16 + rbase + i;
          if (node < nn)
            feats[(size_t)node * OUT_DIM + cw + col] = cu[mt][i] + buv;
        }
      }
    }
  }
}

// -------------------------- K2: per-edge logits + segment-max (atomicMax)
// 4 lanes per edge (one head each) -> 512B contiguous Q/K row gather per edge.
__global__ __launch_bounds__(256) void k_logits(
    const float* __restrict__ Q, const float* __restrict__ Kp,
    const int* __restrict__ senders, const int* __restrict__ receivers,
    float* __restrict__ logits, unsigned* __restrict__ nodeMax, int ne) {
  int t  = blockIdx.x * blockDim.x + threadIdx.x;
  int ed = t >> 2, h = t & 3;
  if (ed >= ne) return;
  int s = senders[ed], r = receivers[ed];
  const float4* q4 = (const float4*)(Q  + (size_t)r * HID + h * OUT_DIM);
  const float4* k4 = (const float4*)(Kp + (size_t)s * HID + h * OUT_DIM);
  float acc = 0.0f;
#pragma unroll
  for (int i = 0; i < OUT_DIM / 4; ++i) {
    float4 qa = q4[i], ka = k4[i];
    acc = fmaf(qa.x, ka.x, acc); acc = fmaf(qa.y, ka.y, acc);
    acc = fmaf(qa.z, ka.z, acc); acc = fmaf(qa.w, ka.w, acc);
  }
  float lg = acc * 0.17677669529663687f;  // 1/sqrt(OUT_DIM)
  logits[(size_t)ed * N_HEADS + h] = lg;
  atomicMax(&nodeMax[(size_t)r * N_HEADS + h], fenc(lg));
}

// -------------------------------- K3: exp(logit - segmax) + segment-sum
__global__ __launch_bounds__(256) void k_expsum(
    const int* __restrict__ receivers, const unsigned* __restrict__ nodeMax,
    float* __restrict__ logits, float* __restrict__ nodeSum, int ne) {
  int t  = blockIdx.x * blockDim.x + threadIdx.x;
  int ed = t >> 2, h = t & 3;
  if (ed >= ne) return;
  int r = receivers[ed];
  float m = fdec(nodeMax[(size_t)r * N_HEADS + h]);
  float e = expf(logits[(size_t)ed * N_HEADS + h] - m);
  logits[(size_t)ed * N_HEADS + h] = e;  // reuse buffer: now holds exp values
  atomicAdd(&nodeSum[(size_t)r * N_HEADS + h], e);
}

// ---- K4: e=edges@We (WMMA f32) fused with attn-weighted message scatter.
// 64 edges/block. Head contributions are reduced in LDS so only one
// atomicAdd per (edge, out-dim) element reaches L2 (4x fewer atomics).
#define LDSE_STRIDE 68   // 64 + 4 pad
#define LDSM_STRIDE 132  // 128 + 4 pad
__global__ __launch_bounds__(256) void k_edge_msg(
    const float* __restrict__ edges, const float* __restrict__ We,
    const int* __restrict__ senders, const int* __restrict__ receivers,
    const float* __restrict__ Vp, const float* __restrict__ expv,
    const float* __restrict__ nodeSum, float* __restrict__ aggr, int ne) {
  __shared__ float Es[16 * MT * LDSE_STRIDE];  // 64 x 68 f32  = 17.4 KB
  __shared__ float Ms[16 * MT * LDSM_STRIDE];  // 64 x 132 f32 = 33.8 KB
  const int tid = threadIdx.x;
  const int eb  = blockIdx.x * (16 * MT);
  const bool full = (eb + 16 * MT) <= ne;      // block-uniform tail test

  // Stage 64x64 fp32 edge tile into LDS: 64B/thread via 4 async b128.
  {
    int r  = tid >> 2;
    int ge = eb + r; if (ge >= ne) ge = ne - 1;
    int cf = (tid & 3) * 16;
    uint32_t lds = (uint32_t)(uintptr_t)(&Es[r * LDSE_STRIDE + cf]);
    unsigned long long g =
        (unsigned long long)(uintptr_t)(edges + (size_t)ge * D_EDGE + cf);
    asm volatile(
        "global_load_async_to_lds_b128 %0, %1, off\n\t"
        "global_load_async_to_lds_b128 %0, %1, off offset:16\n\t"
        "global_load_async_to_lds_b128 %0, %1, off offset:32\n\t"
        "global_load_async_to_lds_b128 %0, %1, off offset:48\n\t"
        "s_wait_asynccnt 0"
        :: "v"(lds), "v"(g) : "memory");
  }
  __syncthreads();

  const int wave = tid >> 5;
  const int lane = tid & 31;
  const int col  = lane & 15;
  const int half = lane >> 4;
  const int ksel = half * 2;
  const int cw   = wave * 16;

  v8f ce[MT] = {};
  for (int kk = 0; kk < D_EDGE / 4; ++kk) {
    const int k0 = kk * 4 + ksel;
    v2f b;
    b.x = We[(size_t)k0 * HID + cw + col];
    b.y = We[(size_t)(k0 + 1) * HID + cw + col];
#pragma unroll
    for (int mt = 0; mt < MT; ++mt) {
      v2f a;
      a.x = Es[(mt * 16 + col) * LDSE_STRIDE + k0];
      a.y = Es[(mt * 16 + col) * LDSE_STRIDE + k0 + 1];
      ce[mt] = wmma4(a, b, ce[mt]);
    }
  }

  // Per-column attention-weighted values -> LDS for the head reduction.
  const int c = cw + col;   // HID column 0..127
  const int h = c >> 5;     // head
  const int rbase = half * 8;
  if (full) {
#pragma unroll
    for (int mt = 0; mt < MT; ++mt) {
#pragma unroll
      for (int i = 0; i < 8; ++i) {
        const int row = mt * 16 + rbase + i;
        const int ed  = eb + row;
        const int s = senders[ed], r = receivers[ed];
        const float attn = expv[(size_t)ed * N_HEADS + h] /
                           nodeSum[(size_t)r * N_HEADS + h];
        Ms[row * LDSM_STRIDE + c] =
            attn * (Vp[(size_t)s * HID + c] + ce[mt][i]);
      }
    }
  } else {
#pragma unroll
    for (int mt = 0; mt < MT; ++mt) {
#pragma unroll
      for (int i = 0; i < 8; ++i) {
        const int row = mt * 16 + rbase + i;
        const int ed  = eb + row;
        float w = 0.0f;
        if (ed < ne) {
          const int s = senders[ed], r = receivers[ed];
          const float attn = expv[(size_t)ed * N_HEADS + h] /
                             nodeSum[(size_t)r * N_HEADS + h];
          w = attn * (Vp[(size_t)s * HID + c] + ce[mt][i]);
        }
        Ms[row * LDSM_STRIDE + c] = w;
      }
    }
  }
  __syncthreads();

  // Reduce the 4 head slices and emit one atomic per (edge, out-dim).
  if (full) {
#pragma unroll
    for (int p = tid; p < 16 * MT * OUT_DIM; p += 256) {
      const int el = p >> 5;       // edge within block
      const int j  = p & 31;       // out-dim index
      const int r  = receivers[eb + el];
      const float* m = &Ms[el * LDSM_STRIDE];
      const float msg = 0.25f * (m[j] + m[j + 32] + m[j + 64] + m[j + 96]);
      atomicAdd(&aggr[(size_t)r * OUT_DIM + j], msg);
    }
  } else {
    for (int p = tid; p < 16 * MT * OUT_DIM; p += 256) {
      const int el = p >> 5;
      const int j  = p & 31;
      const int ed = eb + el;
      if (ed < ne) {
        const int r = receivers[ed];
        const float* m = &Ms[el * LDSM_STRIDE];
        const float msg = 0.25f * (m[j] + m[j + 32] + m[j + 64] + m[j + 96]);
        atomicAdd(&aggr[(size_t)r * OUT_DIM + j], msg);
      }
    }
  }
}

// ------------------------------------------------ K5: relu(feats + aggr)
__global__ __launch_bounds__(256) void k_final(const float* __restrict__ feats,
                                               const float* __restrict__ aggr,
                                               float* __restrict__ out, int n) {
  int t = blockIdx.x * blockDim.x + threadIdx.x;
  if (t < n) {
    float v = feats[t] + aggr[t];
    out[t] = v > 0.0f ? v : 0.0f;
  }
}

// -------------------------------------------------------------------------
extern "C" void kernel_launch(void* const* d_in, const int* in_sizes, int n_in,
                              void* d_out, int out_size, void* d_ws,
                              size_t ws_size, hipStream_t stream) {
  (void)n_in; (void)ws_size; (void)out_size;
  const float* nodes     = (const float*)d_in[0];
  const float* edges     = (const float*)d_in[1];
  const int*   senders   = (const int*)d_in[2];
  const int*   receivers = (const int*)d_in[3];
  const float* Wq = (const float*)d_in[4];
  const float* bq = (const float*)d_in[5];
  const float* Wk = (const float*)d_in[6];
  const float* bk = (const float*)d_in[7];
  const float* Wv = (const float*)d_in[8];
  const float* bv = (const float*)d_in[9];
  const float* We = (const float*)d_in[10];
  const float* Wu = (const float*)d_in[11];
  const float* bu = (const float*)d_in[12];

  const int nn = in_sizes[0] / D_FEAT;  // 50000
  const int ne = in_sizes[2];           // 800000

  // Workspace carve-out (~104 MB total), 256B-aligned slices.
  char* p = (char*)d_ws;
  auto take = [&](size_t bytes) {
    char* r = p;
    p += (bytes + 255) & ~(size_t)255;
    return r;
  };
  float*    Q       = (float*)take((size_t)nn * HID * 4);
  float*    Kp      = (float*)take((size_t)nn * HID * 4);
  float*    Vp      = (float*)take((size_t)nn * HID * 4);
  float*    feats   = (float*)take((size_t)nn * OUT_DIM * 4);
  float*    aggr    = (float*)take((size_t)nn * OUT_DIM * 4);
  float*    logits  = (float*)take((size_t)ne * N_HEADS * 4);
  float*    nodeSum = (float*)take((size_t)nn * N_HEADS * 4);
  unsigned* nodeMax = (unsigned*)take((size_t)nn * N_HEADS * 4);

  const int initN = nn * OUT_DIM;
  const int rowsPerBlk = 16 * MT;
  k_init<<<(initN + 255) / 256, 256, 0, stream>>>(aggr, nodeSum, nodeMax, nn);
  k_node_proj<<<(nn + rowsPerBlk - 1) / rowsPerBlk, 256, 0, stream>>>(
      nodes, Wq, bq, Wk, bk, Wv, bv, Wu, bu, Q, Kp, Vp, feats, nn);
  k_logits<<<(ne * 4 + 255) / 256, 256, 0, stream>>>(Q, Kp, senders, receivers,
                                                     logits, nodeMax, ne);
  k_expsum<<<(ne * 4 + 255) / 256, 256, 0, stream>>>(receivers, nodeMax,
                                                     logits, nodeSum, ne);
  k_edge_msg<<<(ne + rowsPerBlk - 1) / rowsPerBlk, 256, 0, stream>>>(
      edges, We, senders, receivers, Vp, logits, nodeSum, aggr, ne);
  k_final<<<(nn * OUT_DIM + 255) / 256, 256, 0, stream>>>(
      feats, aggr, (float*)d_out, nn * OUT_DIM);
}